// FocusWindow_63247688401523
// MI455X (gfx1250) — compile-verified
//
#include <hip/hip_runtime.h>
#include <hip/hip_bf16.h>
#include <math.h>

typedef float v2f __attribute__((ext_vector_type(2)));
typedef float v8f __attribute__((ext_vector_type(8)));

#define WMMA4(A, B, C) \
  __builtin_amdgcn_wmma_f32_16x16x4_f32(false, (A), false, (B), (short)0, (C), false, false)

constexpr int DMODEL = 1024;
constexpr int NHEAD = 16;
constexpr int HD = DMODEL / NHEAD;      // 64
constexpr int ROWB = DMODEL * 4;        // 4096 bytes per row
constexpr int TILEB = 16 * ROWB;        // 65536 bytes per 16-row K tile
constexpr unsigned SMEM_BYTES = 2u * TILEB;  // double buffer

// ---------------------------------------------------------------- utilities
__global__ void fw_copy(const float* __restrict__ src, float* __restrict__ dst, unsigned long long n) {
  unsigned long long i = (unsigned long long)blockIdx.x * blockDim.x + threadIdx.x;
  if (i < n) dst[i] = src[i];
}

__global__ void fw_zero_slots(float* scal, int n) {
  int i = threadIdx.x;
  if (i < n) scal[i] = 0.f;
}

// mean(f0) -> span_scale -> temperature -> stfac = hd^-0.5 * temperature
__global__ void fw_f0_scalars(const float* __restrict__ f0, int n,
                              const float* __restrict__ temp_scale,
                              float* __restrict__ scal) {
  __shared__ float red[256];
  float s = 0.f;
  for (int i = threadIdx.x; i < n; i += 256) s += f0[i];
  red[threadIdx.x] = s;
  __syncthreads();
  for (int off = 128; off > 0; off >>= 1) {
    if (threadIdx.x < off) red[threadIdx.x] += red[threadIdx.x + off];
    __syncthreads();
  }
  if (threadIdx.x == 0) {
    float mean = red[0] / (float)n;
    float span = 700.f * (__expf(mean / 1127.f) - 1.f);
    float temperature = 1.f + temp_scale[0] * (1.f - span);
    scal[0] = rsqrtf((float)HD) * temperature;
  }
}

// one block per row: atomicAdd(||row||) into slot
__global__ void fw_row_norms(const float* __restrict__ base, float* slot) {
  __shared__ float red[256];
  const float* p = base + (unsigned long long)blockIdx.x * DMODEL;
  float s = 0.f;
  for (int i = threadIdx.x; i < DMODEL; i += 256) { float v = p[i]; s += v * v; }
  red[threadIdx.x] = s;
  __syncthreads();
  for (int off = 128; off > 0; off >>= 1) {
    if (threadIdx.x < off) red[threadIdx.x] += red[threadIdx.x + off];
    __syncthreads();
  }
  if (threadIdx.x == 0) atomicAdd(slot, sqrtf(red[0]));
}

// wslot: [0]=qnorm_sum [1]=knorm_sum [2]=max_iter [3]=active [4]=diffsum
__global__ void fw_finalize_win(float* wslot, float inv_q, float inv_k) {
  if (threadIdx.x == 0) {
    float interest = 0.5f * (wslot[0] * inv_q + wslot[1] * inv_k);
    wslot[2] = fminf(floorf(3.f + interest * 12.f), 20.f);
    wslot[3] = (0.f < wslot[2]) ? 1.f : 0.f;
    wslot[4] = 0.f;
  }
}

__global__ void fw_flags(float* wslot, const float* __restrict__ thr,
                         const float* __restrict__ sf, float invcnt, float next_i) {
  if (threadIdx.x == 0) {
    if (wslot[3] != 0.f) {
      float diff = wslot[4] * invcnt;
      float t = thr[0], s = sf[0];
      bool notconv = diff >= (t + s * diff);
      bool act = notconv && (next_i < wslot[2]);
      wslot[3] = act ? 1.f : 0.f;
    }
    wslot[4] = 0.f;
  }
}

// Async-stage one 16x1024 K tile into LDS buffer `buf` with per-row col4 XOR
// swizzle (keeps both row-fragment and column-fragment LDS reads conflict-free
// with zero padding). 8 x b128 per thread, tracked on ASYNCcnt.
__device__ __forceinline__ void fw_stage_tile(const float* kbase, unsigned smemBase,
                                              int buf, int kb, int eff, int tid) {
#pragma unroll
  for (int i = 0; i < 8; ++i) {
    const int chunk = i * 512 + tid;       // 0..4095 b128 chunks
    const int row = chunk >> 8;            // 0..15
    const int col4 = chunk & 255;          // 0..255 (16B units)
    int kr = kb + row;
    if (kr >= eff) kr = eff - 1;
    const unsigned gOff = (unsigned)kr * (unsigned)ROWB + (unsigned)col4 * 16u;
    const int scol4 = (col4 & ~15) | ((col4 ^ row) & 15);
    const unsigned lOff = smemBase + (unsigned)buf * (unsigned)TILEB +
                          (unsigned)row * (unsigned)ROWB + (unsigned)scol4 * 16u;
    asm volatile("global_load_async_to_lds_b128 %0, %1, %2"
                 :
                 : "v"(lOff), "v"(gOff), "s"(kbase)
                 : "memory");
  }
}

// ----------------------------------------------------- flash-attention step
// grid.x = ceil(eff/16) query tiles; block = 512 = 16 waves, wave h = head h.
// S^T = K*Q^T (softmax row == lane), O^T = K^T * P^T; K tiles double-buffered
// in LDS via async loads.
__global__ __launch_bounds__(512) void fw_attn_step(
    float* __restrict__ qbuf,         // window q rows (read + q+=out)
    const float* __restrict__ kbase,  // K rows (original x slice, fixed)
    float* __restrict__ obase,        // out rows (also serves as prev)
    int eff, const float* __restrict__ scal, const float* __restrict__ wslot,
    float* __restrict__ diffsum, int first) {
  extern __shared__ float fw_smem[];
  if (((const volatile float*)wslot)[3] == 0.f) return;  // uniform early-out
  const float stfac = scal[0];
  const int tid = threadIdx.x;
  const int lane = tid & 31;
  const int head = tid >> 5;
  const int l15 = lane & 15;
  const int g = lane >> 4;
  const int koff = 2 * g;
  const int row0 = blockIdx.x * 16;
  const int myq = row0 + l15;
  const int qr = (myq < eff) ? myq : (eff - 1);
  const float* qrow = qbuf + (unsigned long long)qr * DMODEL + head * HD;
  const unsigned smemBase = (unsigned)(unsigned long long)(uintptr_t)fw_smem;

  const int nkt = (eff + 15) >> 4;
  fw_stage_tile(kbase, smemBase, 0, 0, eff, tid);  // prologue stage

  // Q fragments: B-operand layout, b[j] = Q[q=lane&15][4c + koff + j]
  v2f qa[16];
#pragma unroll
  for (int c = 0; c < 16; ++c) qa[c] = *(const v2f*)(qrow + 4 * c + koff);

  float mrow = -3.0e38f;
  float lsum = 0.f;
  const v8f vzero = {0.f, 0.f, 0.f, 0.f, 0.f, 0.f, 0.f, 0.f};
  v8f o[4];
#pragma unroll
  for (int f = 0; f < 4; ++f) o[f] = vzero;

  for (int kt = 0; kt < nkt; ++kt) {
    if (kt + 1 < nkt) {
      fw_stage_tile(kbase, smemBase, (kt + 1) & 1, (kt + 1) * 16, eff, tid);
      asm volatile("s_wait_asynccnt 0x8" ::: "memory");  // retire tile kt
    } else {
      asm volatile("s_wait_asynccnt 0x0" ::: "memory");
    }
    __syncthreads();  // all waves' stage of tile kt complete
    const float* sm = fw_smem + (unsigned)(kt & 1) * (unsigned)(TILEB / 4);

    // S^T[kv][q]: A = K tile rows (LDS, swizzled), B = Q rows
    v8f st = vzero;
#pragma unroll
    for (int c = 0; c < 16; ++c) {
      const float* p = sm + l15 * DMODEL + head * HD + ((c ^ l15) << 2) + koff;
      v2f ka = *(const v2f*)p;
      st = WMMA4(ka, qa[c], st);
    }

    // scale + mask invalid kv rows; online softmax (query == lane column)
    const int kb = kt * 16;
    float p[8];
    float mloc = -3.0e38f;
#pragma unroll
    for (int v = 0; v < 8; ++v) {
      const int kv = kb + v + 8 * g;
      p[v] = (kv < eff) ? st[v] * stfac : -3.0e38f;
      mloc = fmaxf(mloc, p[v]);
    }
    mloc = fmaxf(mloc, __shfl_xor(mloc, 16, 32));
    const float mnew = fmaxf(mrow, mloc);
    const float alpha = __expf(mrow - mnew);
    float psum = 0.f;
#pragma unroll
    for (int v = 0; v < 8; ++v) {
      p[v] = __expf(p[v] - mnew);
      psum += p[v];
    }
    psum += __shfl_xor(psum, 16, 32);
    lsum = lsum * alpha + psum;
    mrow = mnew;
#pragma unroll
    for (int f = 0; f < 4; ++f) o[f] *= alpha;

    // P^T B-fragments from S^T C-layout: per chunk, kv>=8 iff c>=2 (uniform)
    v2f bp[4];
#pragma unroll
    for (int c = 0; c < 4; ++c) {
      const int src = ((c >= 2) ? 16 : 0) + l15;
      const float lo0 = __shfl(p[(4 * c + 0) & 7], src, 32);
      const float hi0 = __shfl(p[(4 * c + 2) & 7], src, 32);
      const float lo1 = __shfl(p[(4 * c + 1) & 7], src, 32);
      const float hi1 = __shfl(p[(4 * c + 3) & 7], src, 32);
      bp[c][0] = g ? hi0 : lo0;
      bp[c][1] = g ? hi1 : lo1;
    }

    // O^T[ch][q] += K^T chunk * P^T chunk (staged rows already clamped; the
    // duplicated out-of-range rows carry p==0 so they contribute nothing)
#pragma unroll
    for (int c = 0; c < 4; ++c) {
      const int r0 = 4 * c + koff;  // local rows 0..15
      const int r1 = r0 + 1;
#pragma unroll
      for (int f = 0; f < 4; ++f) {
        const int cl = 4 * f + (l15 >> 2);  // local col4 within head slice
        v2f a;
        a[0] = sm[r0 * DMODEL + head * HD + (((cl ^ r0) & 15) << 2) + (l15 & 3)];
        a[1] = sm[r1 * DMODEL + head * HD + (((cl ^ r1) & 15) << 2) + (l15 & 3)];
        o[f] = WMMA4(a, bp[c], o[f]);
      }
    }
    __syncthreads();  // buffer (kt&1) free for restage at kt+2
  }

  const float rl = 1.f / lsum;
  float dacc = 0.f;
  if (myq < eff) {
    float* orow = obase + (unsigned long long)myq * DMODEL + head * HD;
    float* qw = qbuf + (unsigned long long)myq * DMODEL + head * HD;
#pragma unroll
    for (int f = 0; f < 4; ++f) {
      const int chb = 16 * f + 8 * g;  // comps v=0..7 are consecutive channels
      float4 lo = make_float4(o[f][0] * rl, o[f][1] * rl, o[f][2] * rl, o[f][3] * rl);
      float4 hi = make_float4(o[f][4] * rl, o[f][5] * rl, o[f][6] * rl, o[f][7] * rl);
      float4 plo = make_float4(0.f, 0.f, 0.f, 0.f);
      float4 phi = make_float4(0.f, 0.f, 0.f, 0.f);
      if (!first) {
        plo = *(const float4*)(orow + chb);
        phi = *(const float4*)(orow + chb + 4);
      }
      dacc += fabsf(lo.x - plo.x) + fabsf(lo.y - plo.y) + fabsf(lo.z - plo.z) + fabsf(lo.w - plo.w);
      dacc += fabsf(hi.x - phi.x) + fabsf(hi.y - phi.y) + fabsf(hi.z - phi.z) + fabsf(hi.w - phi.w);
      *(float4*)(orow + chb) = lo;
      *(float4*)(orow + chb + 4) = hi;
      float4 ql = *(const float4*)(qw + chb);
      float4 qh = *(const float4*)(qw + chb + 4);
      ql.x += lo.x; ql.y += lo.y; ql.z += lo.z; ql.w += lo.w;
      qh.x += hi.x; qh.y += hi.y; qh.z += hi.z; qh.w += hi.w;
      *(float4*)(qw + chb) = ql;
      *(float4*)(qw + chb + 4) = qh;
    }
  }
#pragma unroll
  for (int off = 16; off >= 1; off >>= 1) dacc += __shfl_xor(dacc, off, 32);
  if (lane == 0) atomicAdd(diffsum, dacc);
}

// ------------------------------------------------------------------- driver
extern "C" void kernel_launch(void* const* d_in, const int* in_sizes, int n_in,
                              void* d_out, int out_size, void* d_ws, size_t ws_size,
                              hipStream_t stream) {
  const float* x = (const float*)d_in[0];
  const float* f0 = (const float*)d_in[1];
  const float* temp_scale = (const float*)d_in[2];
  const float* threshold = (const float*)d_in[3];
  const float* s_factor = (const float*)d_in[4];
  float* out = (float*)d_out;

  const int ctx = in_sizes[0] / DMODEL;  // B == 1
  const int max_dist = in_sizes[1];
  const int WIN = 1000, SPAN = 1200, MAXIT = 20;

  (void)hipFuncSetAttribute((const void*)fw_attn_step,
                            hipFuncAttributeMaxDynamicSharedMemorySize,
                            (int)SMEM_BYTES);

  float* qbuf = (float*)d_ws;
  float* scal = qbuf + (unsigned long long)ctx * DMODEL;

  unsigned long long nq = (unsigned long long)ctx * DMODEL;
  fw_copy<<<dim3((unsigned)((nq + 255) / 256)), 256, 0, stream>>>(x, qbuf, nq);

  // window geometry (mirrors the reference's win-size mutation)
  int wstart[8], wkstart[8], weff[8], wklen[8], wwin[8];
  int wsv = WIN;
  int nwin = (ctx + wsv - 1) / wsv;
  if (nwin > 8) nwin = 8;
  for (int i = 0; i < nwin; ++i) {
    int start = i * wsv;
    int end = (i + 1) * wsv;
    if (end > ctx) end = ctx;
    wsv = end - start;
    int ks = start - SPAN + wsv;
    if (ks < 0) ks = 0;
    int ke = start + SPAN;
    if (ke > ctx) ke = ctx;
    int eff = wsv;
    if (ke - ks < eff) eff = ke - ks;
    if (max_dist < eff) eff = max_dist;
    wstart[i] = start; wkstart[i] = ks; weff[i] = eff; wklen[i] = ke - ks; wwin[i] = wsv;
  }

  fw_zero_slots<<<1, 128, 0, stream>>>(scal, 8 + 8 * nwin);
  fw_f0_scalars<<<1, 256, 0, stream>>>(f0, in_sizes[1], temp_scale, scal);

  for (int w = 0; w < nwin; ++w) {
    float* wslot = scal + 8 + 8 * w;
    fw_row_norms<<<dim3((unsigned)wwin[w]), 256, 0, stream>>>(
        x + (unsigned long long)wstart[w] * DMODEL, wslot + 0);
    fw_row_norms<<<dim3((unsigned)wklen[w]), 256, 0, stream>>>(
        x + (unsigned long long)wkstart[w] * DMODEL, wslot + 1);
    fw_finalize_win<<<1, 32, 0, stream>>>(wslot, 1.f / (float)wwin[w], 1.f / (float)wklen[w]);
  }

  for (int it = 0; it < MAXIT; ++it) {
    for (int w = 0; w < nwin; ++w) {
      float* wslot = scal + 8 + 8 * w;
      const int eff = weff[w];
      const int nqt = (eff + 15) / 16;
      fw_attn_step<<<dim3((unsigned)nqt), 512, SMEM_BYTES, stream>>>(
          qbuf + (unsigned long long)wstart[w] * DMODEL,
          x + (unsigned long long)wkstart[w] * DMODEL,
          out + (unsigned long long)wstart[w] * DMODEL,
          eff, scal, wslot, wslot + 4, it == 0 ? 1 : 0);
      fw_flags<<<1, 32, 0, stream>>>(wslot, threshold, s_factor,
                                     1.f / ((float)eff * (float)DMODEL), (float)(it + 1));
    }
  }
}